// PLabelingNet2Set_41351945126302
// MI455X (gfx1250) — compile-verified
//
#include <hip/hip_runtime.h>
#include <hip/hip_bf16.h>

#define NN 10000
#define FF 64
#define EE 160000
#define MM 8
#define LL 3
#define BB 16
#define TSTRIDE 68   // LDS tile row stride (dwords): bank = (4*row+kk)%64 -> conflict-free

typedef __attribute__((ext_vector_type(2)))  float v2f;
typedef __attribute__((ext_vector_type(8)))  float v8f;

// D = A(16x4 f32) * B(4x16 f32) + C(16x16 f32), full f32 precision (matches reference)
__device__ __forceinline__ v8f wmma4(v2f a, v2f b, v8f c) {
    return __builtin_amdgcn_wmma_f32_16x16x4_f32(
        /*neg_a=*/false, a, /*neg_b=*/false, b,
        /*c_mod=*/(short)0, c, /*reuse_a=*/false, /*reuse_b=*/false);
}

// ---------------------------------------------------------------------------
// sel_id / inv: 16 distinct ids -> rank sort (inv[i] = #ids smaller; sel[rank]=id)
__global__ void k_setup(const int* __restrict__ pos, int* __restrict__ sel,
                        int* __restrict__ invb) {
    int t = threadIdx.x;
    if (t < 2 * MM) {
        int p = pos[t];
        int rank = 0;
        for (int j = 0; j < 2 * MM; ++j) rank += (pos[j] < p) ? 1 : 0;
        invb[t] = rank;
        sel[rank] = p;
    }
}

// ---------------------------------------------------------------------------
// Fused per-layer dense path:  G[b] = ((relu?)X[b] @ W0 + B0) @ WC
// Block = 4 waves; wave w owns col tile 16w. Stage-1 result (16x64) is staged
// in LDS (padded stride) so stage-2 WMMAs see the full K=64 of the tile.
// The 16 label-injected rows of G are fixed up afterwards by k_x1_fix.
template<bool RELU, bool BCAST>
__global__ __launch_bounds__(128)
void k_fused(const float* __restrict__ X, const float* __restrict__ W0,
             const float* __restrict__ B0, const float* __restrict__ WC,
             float* __restrict__ G) {
    __shared__ float tile[16 * TSTRIDE];

    const int lane = threadIdx.x & 31;
    const int wv   = threadIdx.x >> 5;
    const int half = lane >> 4;                // 0: lanes 0-15, 1: lanes 16-31
    const int col  = (wv << 4) + (lane & 15);
    const int rowbase = blockIdx.x * 16;
    const int b = blockIdx.y;

    const float* Xb = BCAST ? X : X + (long)b * ((long)NN * FF);
    const float* arow = Xb + (long)(rowbase + (lane & 15)) * FF;

    // ---- stage 1: t = (relu?)X @ W0 + B0 ----
    v8f acc;
    const float binit = B0[col];
#pragma unroll
    for (int j = 0; j < 8; ++j) acc[j] = binit;

#pragma unroll
    for (int k = 0; k < FF; k += 4) {
        const int kk = k + 2 * half;           // A lane holds K=kk, kk+1
        v2f a = *(const v2f*)(arow + kk);      // 8B aligned (kk even)
        if (RELU) { a.x = fmaxf(a.x, 0.0f); a.y = fmaxf(a.y, 0.0f); }
        v2f bf;                                // B lane holds W0[kk..kk+1][col]
        bf.x = W0[kk * FF + col];
        bf.y = W0[(kk + 1) * FF + col];
        acc = wmma4(a, bf, acc);
    }

    // stage tile: row j+8*half, col  (per-wave conflict-free banks)
#pragma unroll
    for (int j = 0; j < 8; ++j) tile[(j + 8 * half) * TSTRIDE + col] = acc[j];
    __syncthreads();

    // ---- stage 2: g = t @ WC ----
    v8f acc2;
#pragma unroll
    for (int j = 0; j < 8; ++j) acc2[j] = 0.0f;

#pragma unroll
    for (int k = 0; k < FF; k += 4) {
        const int kk = k + 2 * half;
        v2f a = *(const v2f*)(&tile[(lane & 15) * TSTRIDE + kk]);  // ds_load_b64
        v2f bf;
        bf.x = WC[kk * FF + col];
        bf.y = WC[(kk + 1) * FF + col];
        acc2 = wmma4(a, bf, acc2);
    }

    float* Gb = G + (long)b * ((long)NN * FF);
#pragma unroll
    for (int j = 0; j < 8; ++j)
        Gb[(long)(rowbase + j + 8 * half) * FF + col] = acc2[j];
}

// ---------------------------------------------------------------------------
// Fixup for the 16 injected rows:
//   x1 = (relu?)gather16(X) @ W1 + B1;   G[b=r, row=sel[r], :] = x1[r] @ WC
// Single block, 4 waves, same two-stage WMMA structure.
template<bool RELU, bool BCAST>
__global__ __launch_bounds__(128)
void k_x1_fix(const float* __restrict__ X, const float* __restrict__ W1,
              const float* __restrict__ B1, const float* __restrict__ WC,
              const int* __restrict__ sel, float* __restrict__ G) {
    __shared__ float tile[16 * TSTRIDE];

    const int lane = threadIdx.x & 31;
    const int wv   = threadIdx.x >> 5;
    const int half = lane >> 4;
    const int col  = (wv << 4) + (lane & 15);
    const int rr   = lane & 15;                // gathered-row index 0..15

    const int node = sel[rr];
    const float* arow = X + ((BCAST ? 0L : (long)rr * NN) + node) * (long)FF;

    // ---- stage 1: x1 = (relu?)X[gather] @ W1 + B1 ----
    v8f acc;
    const float binit = B1[col];
#pragma unroll
    for (int j = 0; j < 8; ++j) acc[j] = binit;

#pragma unroll
    for (int k = 0; k < FF; k += 4) {
        const int kk = k + 2 * half;
        v2f a = *(const v2f*)(arow + kk);
        if (RELU) { a.x = fmaxf(a.x, 0.0f); a.y = fmaxf(a.y, 0.0f); }
        v2f bf;
        bf.x = W1[kk * FF + col];
        bf.y = W1[(kk + 1) * FF + col];
        acc = wmma4(a, bf, acc);
    }

#pragma unroll
    for (int j = 0; j < 8; ++j) tile[(j + 8 * half) * TSTRIDE + col] = acc[j];
    __syncthreads();

    // ---- stage 2: rows of G = x1 @ WC ----
    v8f acc2;
#pragma unroll
    for (int j = 0; j < 8; ++j) acc2[j] = 0.0f;

#pragma unroll
    for (int k = 0; k < FF; k += 4) {
        const int kk = k + 2 * half;
        v2f a = *(const v2f*)(&tile[(lane & 15) * TSTRIDE + kk]);
        v2f bf;
        bf.x = WC[kk * FF + col];
        bf.y = WC[(kk + 1) * FF + col];
        acc2 = wmma4(a, bf, acc2);
    }

#pragma unroll
    for (int j = 0; j < 8; ++j) {
        const int r = j + 8 * half;            // batch index == gathered row
        G[((long)r * NN + sel[r]) * (long)FF + col] = acc2[j];
    }
}

// ---------------------------------------------------------------------------
__global__ __launch_bounds__(256)
void k_zero(float4* __restrict__ p, long n4) {
    long i = blockIdx.x * (long)blockDim.x + threadIdx.x;
    const long stride = (long)gridDim.x * blockDim.x;
    const float4 z = {0.f, 0.f, 0.f, 0.f};
    for (; i < n4; i += stride) p[i] = z;
}

// ---------------------------------------------------------------------------
// H[b, dst[e], :] += G[b, src[e], :] via global f32 atomics (both buffers L2-resident).
// Thread = (edge, batch, 16-float chunk): 4x float4 load + 16 atomic adds.
__global__ __launch_bounds__(256)
void k_spmm(const float* __restrict__ G, const int* __restrict__ src,
            const int* __restrict__ dst, float* __restrict__ H) {
    const long id = blockIdx.x * (long)blockDim.x + threadIdx.x;
    const long total = (long)EE * BB * 4;
    if (id >= total) return;
    const int  chunk = (int)(id & 3);
    const long t = id >> 2;
    const int  b = (int)(t & (BB - 1));
    const long e = t >> 4;

    const int s = src[e];
    const int d = dst[e];
    const float4* gp = (const float4*)(G + ((long)b * NN + s) * FF + chunk * 16);
    float* hp = H + ((long)b * NN + d) * (long)FF + chunk * 16;
#pragma unroll
    for (int q = 0; q < 4; ++q) {
        const float4 v = gp[q];
        atomicAdd(hp + 4 * q + 0, v.x);
        atomicAdd(hp + 4 * q + 1, v.y);
        atomicAdd(hp + 4 * q + 2, v.z);
        atomicAdd(hp + 4 * q + 3, v.w);
    }
}

// ---------------------------------------------------------------------------
// out[m,r,c,f] = X[inv[m,c]*N + pos[m, r^c], f]   (2048 floats, no relu on last layer)
__global__ void k_gather_out(const float* __restrict__ XF, const int* __restrict__ pos,
                             const int* __restrict__ invb, float* __restrict__ out) {
    const int t = blockIdx.x * blockDim.x + threadIdx.x;
    if (t >= MM * 2 * 2 * FF) return;
    const int f = t & 63;
    const int c = (t >> 6) & 1;
    const int r = (t >> 7) & 1;
    const int m = t >> 8;
    const int batch = invb[m * 2 + c];
    const int node  = pos[m * 2 + (r ^ c)];
    out[t] = XF[((long)batch * NN + node) * (long)FF + f];
}

// ---------------------------------------------------------------------------
extern "C" void kernel_launch(void* const* d_in, const int* in_sizes, int n_in,
                              void* d_out, int out_size, void* d_ws, size_t ws_size,
                              hipStream_t stream) {
    (void)in_sizes; (void)n_in; (void)out_size; (void)ws_size;
    const float* x      = (const float*)d_in[0];
    const int*   src    = (const int*)d_in[1];
    const int*   dst    = (const int*)d_in[2];
    const int*   pos    = (const int*)d_in[3];
    const float* f0_w   = (const float*)d_in[4];
    const float* f0_b   = (const float*)d_in[5];
    const float* f1_w   = (const float*)d_in[6];
    const float* f1_b   = (const float*)d_in[7];
    const float* conv_w = (const float*)d_in[8];
    float* out = (float*)d_out;

    const long TEN = (long)BB * NN * FF;          // 10,240,000 floats (41 MB)
    float* buf0 = (float*)d_ws;                   // g buffer
    float* buf1 = buf0 + TEN;                     // h / x buffer
    int*   sel  = (int*)(buf1 + TEN);
    int*   invb = sel + 2 * MM;

    k_setup<<<1, 32, 0, stream>>>(pos, sel, invb);

    const dim3 ggrid(NN / 16, BB);                // 625 x 16 row tiles
    const long spmmThreads = (long)EE * BB * 4;   // 10.24M

    for (int i = 0; i < LL; ++i) {
        const float* W0 = f0_w + (long)i * FF * FF;
        const float* B0 = f0_b + (long)i * FF;
        const float* W1 = f1_w + (long)i * FF * FF;
        const float* B1 = f1_b + (long)i * FF;
        const float* WC = conv_w + (long)i * FF * FF;

        // g = ((relu?)X @ W0 + B0) @ WC                      -> buf0
        // then fix the 16 injected rows: g_row = x1 @ WC     -> buf0
        if (i == 0) {
            k_fused<false, true><<<ggrid, 128, 0, stream>>>(x, W0, B0, WC, buf0);
            k_x1_fix<false, true><<<1, 128, 0, stream>>>(x, W1, B1, WC, sel, buf0);
        } else {
            k_fused<true, false><<<ggrid, 128, 0, stream>>>(buf1, W0, B0, WC, buf0);
            k_x1_fix<true, false><<<1, 128, 0, stream>>>(buf1, W1, B1, WC, sel, buf0);
        }

        // h = segment_sum(g[src], dst)                       -> buf1
        k_zero<<<2048, 256, 0, stream>>>((float4*)buf1, TEN / 4);
        k_spmm<<<(int)((spmmThreads + 255) / 256), 256, 0, stream>>>(buf0, src, dst, buf1);
        // ReLU folded into the next layer's A-fragment loads (none after last layer)
    }

    k_gather_out<<<8, 256, 0, stream>>>(buf1, pos, invb, out);
}